// Cutmix_63376537420167
// MI455X (gfx1250) — compile-verified
//
#include <hip/hip_runtime.h>

// Problem constants (match reference).
#define B_ 128
#define L_ 160000
#define C_ 128

typedef __attribute__((ext_vector_type(2))) float v2f;
typedef __attribute__((ext_vector_type(8))) float v8f;

// ---------------------------------------------------------------------------
// Kernel 1: wave_mix — bandwidth-bound streaming select.
// One float4 per thread; grid.y = batch row. Region classification per chunk
// avoids reading both source rows except at the (rare) crop boundaries.
// ---------------------------------------------------------------------------
__global__ __launch_bounds__(256) void cutmix_wave_kernel(
    const float* __restrict__ wave, const float* __restrict__ lam,
    const int* __restrict__ perm, const float* __restrict__ dec,
    const int* __restrict__ start, float* __restrict__ out)
{
    const int b  = blockIdx.y;                 // uniform -> SGPR metadata loads
    const int i4 = blockIdx.x * 256 + threadIdx.x;
    const int n4 = L_ / 4;                     // 40000 float4 per row
    if (i4 >= n4) return;

    const float lam_b = lam[b];
    const int   crop  = (int)((1.0f - lam_b) * (float)L_);  // matches jnp astype(int32)
    const int   s     = start[b];
    const int   e     = s + crop;
    const bool  d     = (dec[b] == 1.0f);
    const int   p     = perm[b];

    const float4* rowA = (const float4*)(wave + (size_t)b * L_);
    float4*       orow = (float4*)(out + (size_t)b * L_);

    const int p0 = i4 * 4;
    float4 v;
    if (!d || e <= p0 || s >= p0 + 4) {
        // chunk entirely outside crop (or no mixing for this row)
        v = rowA[i4];
    } else {
        const float4* rowB = (const float4*)(wave + (size_t)p * L_);
        if (s <= p0 && p0 + 4 <= e) {
            // chunk entirely inside crop
            v = rowB[i4];
        } else {
            // boundary chunk: per-element select
            float4 a  = rowA[i4];
            float4 bb = rowB[i4];
            v.x = (p0 + 0 >= s && p0 + 0 < e) ? bb.x : a.x;
            v.y = (p0 + 1 >= s && p0 + 1 < e) ? bb.y : a.y;
            v.z = (p0 + 2 >= s && p0 + 2 < e) ? bb.z : a.z;
            v.w = (p0 + 3 >= s && p0 + 3 < e) ? bb.w : a.w;
        }
    }
    orow[i4] = v;
}

// ---------------------------------------------------------------------------
// Kernel 2: label_out = M @ onehot via V_WMMA_F32_16X16X4_F32, where
//   M = diag(a) + diag(c) * P,  a = dec*lam + (1-dec),  c = dec*(1-lam).
// One wave (32 threads) per 16x16 output tile; K=128 in steps of 4.
// A-fragment of M is synthesized per lane (<=2 nonzeros per row) — M is never
// materialized. EXEC is all-1s (blockDim == 32, no divergence before WMMA).
// ---------------------------------------------------------------------------
__global__ __launch_bounds__(32) void cutmix_label_wmma_kernel(
    const float* __restrict__ onehot, const float* __restrict__ lam,
    const int* __restrict__ perm, const float* __restrict__ dec,
    float* __restrict__ label_out)
{
    const int tile   = blockIdx.x;   // 0..63 over 8x8 tiles of the 128x128 output
    const int ti     = tile >> 3;
    const int tj     = tile & 7;
    const int l      = threadIdx.x;  // 0..31
    const int half   = l >> 4;       // lane-half selects K offset {0,1} vs {2,3}
    const int lane16 = l & 15;

    // This lane supplies A-matrix row r for all K.
    const int   r      = ti * 16 + lane16;
    const float lam_r  = lam[r];
    const bool  d_r    = (dec[r] == 1.0f);
    const int   perm_r = perm[r];
    const float a_r    = d_r ? lam_r : 1.0f;          // weight on onehot[r]
    const float c_r    = d_r ? (1.0f - lam_r) : 0.0f; // weight on onehot[perm[r]]

    // This lane supplies B-matrix column n for all K.
    const int n = tj * 16 + lane16;

    v8f acc = {};
    for (int k = 0; k < C_; k += 4) {
        const int kb = k + half * 2;  // A/B (lane,vgpr)->K mapping: v0=K kb, v1=K kb+1
        v2f A, Bv;
        A.x = ((kb     == r) ? a_r : 0.0f) + ((kb     == perm_r) ? c_r : 0.0f);
        A.y = ((kb + 1 == r) ? a_r : 0.0f) + ((kb + 1 == perm_r) ? c_r : 0.0f);
        Bv.x = onehot[kb * C_ + n];
        Bv.y = onehot[(kb + 1) * C_ + n];
        // (neg_a, A, neg_b, B, c_mod, C, reuse_a, reuse_b)
        acc = __builtin_amdgcn_wmma_f32_16x16x4_f32(
            false, A, false, Bv, (short)0, acc, false, false);
    }

    // C/D layout: VGPR v -> row ti*16 + v + (lane>=16 ? 8 : 0), col n.
#pragma unroll
    for (int vv = 0; vv < 8; ++vv) {
        const int ro = ti * 16 + vv + half * 8;
        label_out[ro * C_ + n] = acc[vv];
    }
}

// ---------------------------------------------------------------------------
// Launch: outputs are concatenated [wave_mix (B*L f32), label_out (B*C f32)].
// Inputs in setup_inputs() order: wave, onehot, lam, perm, dec, start.
// ---------------------------------------------------------------------------
extern "C" void kernel_launch(void* const* d_in, const int* in_sizes, int n_in,
                              void* d_out, int out_size, void* d_ws, size_t ws_size,
                              hipStream_t stream)
{
    const float* wave   = (const float*)d_in[0];
    const float* onehot = (const float*)d_in[1];
    const float* lam    = (const float*)d_in[2];
    const int*   perm   = (const int*)d_in[3];
    const float* dec    = (const float*)d_in[4];
    const int*   start  = (const int*)d_in[5];

    float* out_wave  = (float*)d_out;
    float* out_label = (float*)d_out + (size_t)B_ * L_;

    dim3 grid((L_ / 4 + 255) / 256, B_);
    cutmix_wave_kernel<<<grid, 256, 0, stream>>>(wave, lam, perm, dec, start, out_wave);

    cutmix_label_wmma_kernel<<<64, 32, 0, stream>>>(onehot, lam, perm, dec, out_label);
}